// GuidedSpatiallyVaryingConv_86938728005899
// MI455X (gfx1250) — compile-verified
//
#include <hip/hip_runtime.h>
#include <math.h>

typedef __attribute__((ext_vector_type(2))) float v2f;
typedef __attribute__((ext_vector_type(8))) float v8f;

#define HH 512
#define WW 512
#define NPIX (HH * WW)

// ---------------------------------------------------------------------------
// 3x3 conv (pad=1) + bias + ReLU via V_WMMA_F32_16X16X4_F32, fully unrolled.
// Implicit GEMM with K ordered kpos-major / channel-minor, channels padded to
// CINP (multiple of 4, zero-filled) so every lane's K-pair is two adjacent
// channels -> single 8-byte LDS loads with immediate offsets.
// Block = 128 threads = 4 waves; each wave owns 16 pixels of a 64-pixel row
// segment. COUT <= 32 via dual accumulators sharing the B fragment.
// ---------------------------------------------------------------------------
template <int CIN0, int CIN1, int COUT>
__global__ __launch_bounds__(128) void conv3x3_wmma(
    const float* __restrict__ in0,
    const float* __restrict__ in1,
    const float* __restrict__ w, const float* __restrict__ bias,
    float* __restrict__ out)
{
  constexpr int CIN  = CIN0 + CIN1;
  constexpr int CINP = (CIN + 3) & ~3;   // padded channels (zeros)
  constexpr int KP   = 9 * CINP;         // padded GEMM K

  __shared__ float halo[198 * CINP];     // [(r*66 + col)*CINP + ci], cols x0-1..x0+64
  __shared__ float wlds[COUT * KP];      // [cout*KP + kpos*CINP + ci]

  const int tid = threadIdx.x;
  const int x0  = blockIdx.x * 64;
  const int y   = blockIdx.y;
  const int b   = blockIdx.z;

  // stage weights: OIHW -> [cout][kpos][ci], zero-pad ci >= CIN
  for (int t = tid; t < COUT * KP; t += 128) {
    int co = t / KP, k = t % KP;
    int kpos = k / CINP, ci = k % CINP;
    wlds[t] = (ci < CIN) ? w[(co * CIN + ci) * 9 + kpos] : 0.0f;
  }
  // stage halo tile: rc = r*66 + col fastest for coalesced global reads
  for (int t = tid; t < CINP * 198; t += 128) {
    int ci = t / 198, rc = t % 198;
    int r = rc / 66, c = rc % 66;
    int gy = y + r - 1, gx = x0 + c - 1;
    float v = 0.0f;
    if (ci < CIN && gy >= 0 && gy < HH && gx >= 0 && gx < WW) {
      v = (ci < CIN0) ? in0[((size_t)(b * CIN0 + ci) * HH + gy) * WW + gx]
                      : in1[((size_t)(b * CIN1 + (ci - CIN0)) * HH + gy) * WW + gx];
    }
    halo[rc * CINP + ci] = v;
  }
  __syncthreads();

  const int lane = tid & 31;
  const int wv   = tid >> 5;       // wave in block: 0..3
  const int half = lane >> 4;      // K-pair select: {k0,k0+1} vs {k0+2,k0+3}
  const int idx  = lane & 15;      // A: M (cout); B: N (pixel)
  const int px   = wv * 16 + idx;  // pixel within 64-wide tile

  // loop-invariant per-lane LDS bases; per-iteration offsets are constants
  const float* aP  = &wlds[idx * KP + 2 * half];
  const float* a2P = &wlds[(idx + 16) * KP + 2 * half];
  const float* bP  = &halo[px * CINP + 2 * half];

  v8f acc0 = {};
  v8f acc1 = {};
#pragma unroll
  for (int k0 = 0; k0 < KP; k0 += 4) {
    const int kpos = k0 / CINP;              // constant per unrolled iter
    const int r    = kpos / 3;
    const int dc   = kpos % 3;
    const int boff = (r * 66 + dc) * CINP + (k0 % CINP);
    v2f a  = *(const v2f*)(aP + k0);
    v2f bv = *(const v2f*)(bP + boff);
    acc0 = __builtin_amdgcn_wmma_f32_16x16x4_f32(
        false, a, false, bv, (short)0, acc0, false, false);
    if constexpr (COUT > 16) {
      v2f a2 = *(const v2f*)(a2P + k0);
      acc1 = __builtin_amdgcn_wmma_f32_16x16x4_f32(
          false, a2, false, bv, (short)0, acc1, false, false);
    }
  }

  const int gx = x0 + px;
#pragma unroll
  for (int rr = 0; rr < 8; ++rr) {
    int co = rr + 8 * half;          // C/D layout: vgpr rr, lane half -> M = rr + 8*half
    float v = acc0[rr] + bias[co];
    v = fmaxf(v, 0.0f);
    out[((size_t)(b * COUT + co) * HH + y) * WW + gx] = v;
    if constexpr (COUT > 16) {
      int co2 = co + 16;
      float v2 = acc1[rr] + bias[co2];
      v2 = fmaxf(v2, 0.0f);
      out[((size_t)(b * COUT + co2) * HH + y) * WW + gx] = v2;
    }
  }
}

// argmax over 19 channels -> float map
__global__ __launch_bounds__(256) void argmax_ch(
    const float* __restrict__ x, float* __restrict__ amap, int total)
{
  int i = blockIdx.x * blockDim.x + threadIdx.x;
  if (i >= total) return;
  int b = i / NPIX, p = i % NPIX;
  const float* xp = x + (size_t)b * 19 * NPIX + p;
  float best = xp[0];
  int bi = 0;
  for (int c = 1; c < 19; ++c) {
    float v = xp[(size_t)c * NPIX];
    if (v > best) { best = v; bi = c; }   // strict > keeps first index (jnp.argmax)
  }
  amap[i] = (float)bi;
}

// 3x3 laplacian (all ones, center -8), zero pad
__global__ __launch_bounds__(256) void laplacian(
    const float* __restrict__ amap, float* __restrict__ edge, int total)
{
  int i = blockIdx.x * blockDim.x + threadIdx.x;
  if (i >= total) return;
  int b = i / NPIX, p = i % NPIX;
  int y = p / WW, x = p % WW;
  const float* a = amap + (size_t)b * NPIX;
  float s = 0.0f;
  for (int dy = -1; dy <= 1; ++dy)
    for (int dx = -1; dx <= 1; ++dx) {
      int yy = y + dy, xx = x + dx;
      if (yy >= 0 && yy < HH && xx >= 0 && xx < WW) s += a[yy * WW + xx];
    }
  edge[i] = s - 9.0f * a[y * WW + x];
}

// 1x1 conv 16 -> 19 (intra head, no relu)
__global__ __launch_bounds__(256) void conv1x1_intra(
    const float* __restrict__ f2, const float* __restrict__ w,
    const float* __restrict__ bias, float* __restrict__ intra, int total)
{
  int i = blockIdx.x * blockDim.x + threadIdx.x;
  if (i >= total) return;
  int b = i / NPIX, p = i % NPIX;
  const float* fp = f2 + (size_t)b * 16 * NPIX + p;
  float fin[16];
#pragma unroll
  for (int c = 0; c < 16; ++c) fin[c] = fp[(size_t)c * NPIX];
  float* op = intra + (size_t)b * 19 * NPIX + p;
  for (int co = 0; co < 19; ++co) {
    float s = bias[co];
#pragma unroll
    for (int ci = 0; ci < 16; ++ci) s += w[co * 16 + ci] * fin[ci];
    op[(size_t)co * NPIX] = s;
  }
}

// 1x1 conv 32 -> 11, sigmoid, normalize across the 11 channels
__global__ __launch_bounds__(256) void guide_head(
    const float* __restrict__ g2, const float* __restrict__ w,
    const float* __restrict__ bias, float* __restrict__ guide, int total)
{
  int i = blockIdx.x * blockDim.x + threadIdx.x;
  if (i >= total) return;
  int b = i / NPIX, p = i % NPIX;
  const float* gp = g2 + (size_t)b * 32 * NPIX + p;
  float gin[32];
#pragma unroll
  for (int c = 0; c < 32; ++c) gin[c] = gp[(size_t)c * NPIX];
  float gv[11];
  float sum = 0.0f;
  for (int co = 0; co < 11; ++co) {
    float s = bias[co];
#pragma unroll
    for (int ci = 0; ci < 32; ++ci) s += w[co * 32 + ci] * gin[ci];
    s = 1.0f / (1.0f + expf(-s));
    gv[co] = s;
    sum += s;
  }
  float inv = 1.0f / (sum + 1e-9f);
  float* op = guide + (size_t)b * 11 * NPIX + p;
  for (int co = 0; co < 11; ++co) op[(size_t)co * NPIX] = gv[co] * inv;
}

// out[c] = g0*x[c] + sum_{t=0..7} g[1+t]*shift_t(x[c]) + g[9]*0 + g[10]*intra[c]
__global__ __launch_bounds__(256) void combine(
    const float* __restrict__ x, const float* __restrict__ intra,
    const float* __restrict__ guide, float* __restrict__ out, int total)
{
  int i = blockIdx.x * blockDim.x + threadIdx.x;
  if (i >= total) return;
  int b   = i / (19 * NPIX);
  int rem = i % (19 * NPIX);
  int c = rem / NPIX, p = rem % NPIX;
  int y = p / WW, xx = p % WW;
  const float* g  = guide + (size_t)b * 11 * NPIX + p;
  const float* xc = x + ((size_t)b * 19 + c) * NPIX;
  float res = g[0] * xc[p];
  int t = 0;
  for (int di = -1; di <= 1; ++di)
    for (int dj = -1; dj <= 1; ++dj) {
      if (di == 0 && dj == 0) continue;   // shift kernel order: (i,j) row-major, skip center
      int yy = y + di, x2 = xx + dj;
      float v = (yy >= 0 && yy < HH && x2 >= 0 && x2 < WW) ? xc[yy * WW + x2] : 0.0f;
      res += g[(size_t)(1 + t) * NPIX] * v;
      ++t;
    }
  res += g[(size_t)10 * NPIX] * intra[((size_t)b * 19 + c) * NPIX + p];
  out[i] = res;
}

extern "C" void kernel_launch(void* const* d_in, const int* in_sizes, int n_in,
                              void* d_out, int out_size, void* d_ws, size_t ws_size,
                              hipStream_t stream)
{
  const float* x     = (const float*)d_in[0];
  const float* image = (const float*)d_in[1];
  const float* iw1   = (const float*)d_in[2];
  const float* ib1   = (const float*)d_in[3];
  const float* iw2   = (const float*)d_in[4];
  const float* ib2   = (const float*)d_in[5];
  const float* iw3   = (const float*)d_in[6];
  const float* ib3   = (const float*)d_in[7];
  const float* gw1   = (const float*)d_in[8];
  const float* gb1   = (const float*)d_in[9];
  const float* gw2   = (const float*)d_in[10];
  const float* gb2   = (const float*)d_in[11];
  const float* gw3   = (const float*)d_in[12];
  const float* gb3   = (const float*)d_in[13];

  const int B = 2;
  // outputs concatenated: out(2*19*NPIX) | guide_feat(2*11*NPIX) | edge(2*NPIX)
  float* outp  = (float*)d_out;
  float* guide = outp + (size_t)B * 19 * NPIX;
  float* edge  = guide + (size_t)B * 11 * NPIX;

  float* ws    = (float*)d_ws;
  float* f1    = ws;                              // 2*16*NPIX
  float* f2    = f1 + (size_t)B * 16 * NPIX;      // 2*16*NPIX
  float* g1    = ws;                              // reuses f1+f2 (2*32*NPIX), safe after intra
  float* intra = f2 + (size_t)B * 16 * NPIX;      // 2*19*NPIX
  float* g2    = intra + (size_t)B * 19 * NPIX;   // 2*32*NPIX
  float* amap  = g2 + (size_t)B * 32 * NPIX;      // 2*NPIX

  dim3 cgrid(WW / 64, HH, B);
  dim3 cblk(128);

  // intra network
  conv3x3_wmma<3, 0, 16><<<cgrid, cblk, 0, stream>>>(image, nullptr, iw1, ib1, f1);
  conv3x3_wmma<16, 0, 16><<<cgrid, cblk, 0, stream>>>(f1, nullptr, iw2, ib2, f2);
  int tot1 = B * NPIX;
  conv1x1_intra<<<(tot1 + 255) / 256, 256, 0, stream>>>(f2, iw3, ib3, intra, tot1);

  // edge path
  argmax_ch<<<(tot1 + 255) / 256, 256, 0, stream>>>(x, amap, tot1);
  laplacian<<<(tot1 + 255) / 256, 256, 0, stream>>>(amap, edge, tot1);

  // guide network on concat([intra(19), edge(1)])
  conv3x3_wmma<19, 1, 32><<<cgrid, cblk, 0, stream>>>(intra, edge, gw1, gb1, g1);
  conv3x3_wmma<32, 0, 32><<<cgrid, cblk, 0, stream>>>(g1, nullptr, gw2, gb2, g2);
  guide_head<<<(tot1 + 255) / 256, 256, 0, stream>>>(g2, gw3, gb3, guide, tot1);

  // final spatially-varying blend
  int tot19 = B * 19 * NPIX;
  combine<<<(tot19 + 255) / 256, 256, 0, stream>>>(x, intra, guide, outp, tot19);
}